// GIN_77249281786393
// MI455X (gfx1250) — compile-verified
//
#include <hip/hip_runtime.h>

// GIN forward for MI455X (gfx1250, wave32).
// Pipeline: pre_mp GEMM (K=21, VALU) -> 3x [scatter-add + WMMA MLP] -> pool -> head.
// GEMMs use V_WMMA_F32_16X16X4_F32 (fp32 in/out, full accuracy, matrix pipe).
// Scatter uses native global_atomic_add_f32 (via unsafeAtomicAdd -> no CAS loop);
// agg (25.6 MB) is L2-resident (192 MB L2), so atomics resolve on-chip.

typedef __attribute__((ext_vector_type(2))) float v2f;
typedef __attribute__((ext_vector_type(8))) float v8f;

#define NN   100000
#define NE   1200000
#define NG   512
#define DIN  21
#define DH   64
#define DOUT 6
#define ROW  68   // padded LDS row stride (floats): bank = (4r + c) % 64 -> conflict-free

// ---------------------------------------------------------------- zero fill
__global__ __launch_bounds__(256) void zero_f4(float4* __restrict__ p, int n4) {
  int i = blockIdx.x * blockDim.x + threadIdx.x;
  if (i < n4) p[i] = make_float4(0.f, 0.f, 0.f, 0.f);
}

// ---------------------------------------------------------------- pre_mp: h = x @ W_pre + b_pre
// 256 threads = 4 nodes x 64 output columns. K=21 is too ragged for WMMA; VALU is fine
// (268 MFLOP total, bound by the 25.6 MB write of h).
__global__ __launch_bounds__(256) void pre_mp_kernel(const float* __restrict__ x,
                                                     const float* __restrict__ W,
                                                     const float* __restrict__ b,
                                                     float* __restrict__ h) {
  __shared__ float sx[4 * DIN];
  const int tid = threadIdx.x;
  const int nb  = blockIdx.x * 4;
  if (tid < 4 * DIN) sx[tid] = x[nb * DIN + tid];
  __syncthreads();
  const int m = tid >> 6;   // node within block
  const int j = tid & 63;   // output column
  float acc = b[j];
  const float* xr = &sx[m * DIN];
#pragma unroll
  for (int k = 0; k < DIN; ++k) acc += xr[k] * W[k * DH + j];
  h[(nb + m) * DH + j] = acc;
}

// ---------------------------------------------------------------- scatter: agg[dst] += h[src]
// 16 lanes per edge, float4 vector read of the message, 4 native fp32 atomics.
__global__ __launch_bounds__(256) void scatter_add_kernel(const float* __restrict__ h,
                                                          const int* __restrict__ ei,
                                                          float* __restrict__ agg) {
  long long t = (long long)blockIdx.x * blockDim.x + threadIdx.x;
  int e  = (int)(t >> 4);
  int c4 = ((int)t & 15) * 4;
  if (e >= NE) return;
  int src = ei[e];
  int dst = ei[NE + e];
  const float4 v = *(const float4*)(h + (long long)src * DH + c4);
  float* ap = agg + (long long)dst * DH + c4;
  unsafeAtomicAdd(ap + 0, v.x);   // -> global_atomic_add_f32 (no CAS loop)
  unsafeAtomicAdd(ap + 1, v.y);
  unsafeAtomicAdd(ap + 2, v.z);
  unsafeAtomicAdd(ap + 3, v.w);
}

// ---------------------------------------------------------------- GIN MLP via WMMA f32 16x16x4
// Block = 128 threads (4 wave32s) handles 16 nodes.
//   A_lds = (h+agg)[16x64], W1/W2 staged in LDS (~41 KB total).
//   Each wave owns 16 output columns; K-loop of 16 x v_wmma_f32_16x16x4_f32 per GEMM.
// Fragment layout (ISA 7.12.2, 32-bit 16x4 A): lane l: r=l&15, hi=l>>4,
//   A frag = {A[r][k+2hi], A[r][k+2hi+1]}, B frag = {W[k+2hi][n], W[k+2hi+1][n]} with n=r,
//   C/D: vgpr i -> row M = 8*hi + i, col = r.
__global__ __launch_bounds__(128) void gin_mlp_kernel(float* __restrict__ h,
                                                      const float* __restrict__ agg,
                                                      const float* __restrict__ W1,
                                                      const float* __restrict__ b1,
                                                      const float* __restrict__ W2,
                                                      const float* __restrict__ b2) {
  __shared__ float A[16 * ROW];
  __shared__ float Z[16 * ROW];
  __shared__ float W1s[DH * DH];
  __shared__ float W2s[DH * DH];

  const int tid  = threadIdx.x;
  const int lane = tid & 31;
  const int wid  = tid >> 5;      // wave 0..3
  const int r    = lane & 15;
  const int hi   = lane >> 4;
  const int base = blockIdx.x * 16;

  // Stage A = h + agg (16x64) -- vectorized float4 loads.
  for (int idx = tid; idx < 16 * (DH / 4); idx += 128) {
    int m = idx >> 4;
    int c = (idx & 15) * 4;
    const float4 hv = *(const float4*)(h   + (long long)(base + m) * DH + c);
    const float4 av = *(const float4*)(agg + (long long)(base + m) * DH + c);
    A[m * ROW + c + 0] = hv.x + av.x;
    A[m * ROW + c + 1] = hv.y + av.y;
    A[m * ROW + c + 2] = hv.z + av.z;
    A[m * ROW + c + 3] = hv.w + av.w;
  }
  // Stage both weight matrices.
  for (int idx = tid; idx < (DH * DH) / 4; idx += 128) {
    ((float4*)W1s)[idx] = ((const float4*)W1)[idx];
    ((float4*)W2s)[idx] = ((const float4*)W2)[idx];
  }
  __syncthreads();

  const int nb = wid * 16;  // this wave's output-column base

  // GEMM1: Z = relu(A @ W1 + b1)
  v8f acc = {};
#pragma unroll
  for (int k = 0; k < DH; k += 4) {
    v2f a, b;
    a.x = A[r * ROW + k + 2 * hi];
    a.y = A[r * ROW + k + 2 * hi + 1];
    b.x = W1s[(k + 2 * hi) * DH + nb + r];
    b.y = W1s[(k + 2 * hi + 1) * DH + nb + r];
    acc = __builtin_amdgcn_wmma_f32_16x16x4_f32(false, a, false, b, (short)0, acc,
                                                false, false);
  }
  {
    const float bias = b1[nb + r];
#pragma unroll
    for (int i = 0; i < 8; ++i) {
      float v = acc[i] + bias;
      Z[(hi * 8 + i) * ROW + nb + r] = v > 0.f ? v : 0.f;
    }
  }
  __syncthreads();

  // GEMM2: h_out = relu(Z @ W2 + b2)  (in place: block owns these 16 rows)
  v8f acc2 = {};
#pragma unroll
  for (int k = 0; k < DH; k += 4) {
    v2f a, b;
    a.x = Z[r * ROW + k + 2 * hi];
    a.y = Z[r * ROW + k + 2 * hi + 1];
    b.x = W2s[(k + 2 * hi) * DH + nb + r];
    b.y = W2s[(k + 2 * hi + 1) * DH + nb + r];
    acc2 = __builtin_amdgcn_wmma_f32_16x16x4_f32(false, a, false, b, (short)0, acc2,
                                                 false, false);
  }
  // Epilogue: bias + ReLU into LDS (reuse A), then fully-coalesced float4 stores.
  {
    const float bias = b2[nb + r];
#pragma unroll
    for (int i = 0; i < 8; ++i) {
      float v = acc2[i] + bias;
      A[(hi * 8 + i) * ROW + nb + r] = v > 0.f ? v : 0.f;
    }
  }
  __syncthreads();
  for (int idx = tid; idx < 16 * (DH / 4); idx += 128) {
    int m = idx >> 4;
    int c = (idx & 15) * 4;
    float4 v = make_float4(A[m * ROW + c + 0], A[m * ROW + c + 1],
                           A[m * ROW + c + 2], A[m * ROW + c + 3]);
    *(float4*)(h + (long long)(base + m) * DH + c) = v;
  }
}

// ---------------------------------------------------------------- pool: g[batch[n]] += h[n]
__global__ __launch_bounds__(256) void pool_kernel(const float* __restrict__ h,
                                                   const int* __restrict__ batch,
                                                   float* __restrict__ g) {
  long long t = (long long)blockIdx.x * blockDim.x + threadIdx.x;
  if (t >= (long long)NN * DH) return;
  int n = (int)(t >> 6);
  int c = (int)t & 63;
  unsafeAtomicAdd(g + (long long)batch[n] * DH + c, h[t]);
}

// ---------------------------------------------------------------- head: relu(g@Wp1+b)@Wp2+b, log_softmax
__global__ __launch_bounds__(64) void post_mp_kernel(const float* __restrict__ g,
                                                     const float* __restrict__ W1,
                                                     const float* __restrict__ b1,
                                                     const float* __restrict__ W2,
                                                     const float* __restrict__ b2,
                                                     float* __restrict__ out) {
  __shared__ float sg[DH];
  __shared__ float st[DH];
  __shared__ float so[DOUT];
  const int gi = blockIdx.x;
  const int j  = threadIdx.x;
  sg[j] = g[gi * DH + j];
  __syncthreads();
  float acc = b1[j];
#pragma unroll 8
  for (int k = 0; k < DH; ++k) acc += sg[k] * W1[k * DH + j];
  st[j] = acc > 0.f ? acc : 0.f;
  __syncthreads();
  if (j < DOUT) {
    float o = b2[j];
#pragma unroll 8
    for (int k = 0; k < DH; ++k) o += st[k] * W2[k * DOUT + j];
    so[j] = o;
  }
  __syncthreads();
  if (j < DOUT) {
    float m = so[0];
#pragma unroll
    for (int k = 1; k < DOUT; ++k) m = fmaxf(m, so[k]);
    float s = 0.f;
#pragma unroll
    for (int k = 0; k < DOUT; ++k) s += __expf(so[k] - m);
    out[gi * DOUT + j] = so[j] - m - logf(s);
  }
}

// ---------------------------------------------------------------- launch
extern "C" void kernel_launch(void* const* d_in, const int* in_sizes, int n_in,
                              void* d_out, int out_size, void* d_ws, size_t ws_size,
                              hipStream_t stream) {
  const float* x     = (const float*)d_in[0];
  const int*   ei    = (const int*)d_in[1];    // [2, E]: src row, then dst row
  const int*   batch = (const int*)d_in[2];
  const float* Wpre  = (const float*)d_in[3];
  const float* bpre  = (const float*)d_in[4];
  const float* Wc1   = (const float*)d_in[5];  // [3,64,64]
  const float* bc1   = (const float*)d_in[6];  // [3,64]
  const float* Wc2   = (const float*)d_in[7];
  const float* bc2   = (const float*)d_in[8];
  const float* Wp1   = (const float*)d_in[9];
  const float* bp1   = (const float*)d_in[10];
  const float* Wp2   = (const float*)d_in[11];
  const float* bp2   = (const float*)d_in[12];
  float* out = (float*)d_out;

  // Workspace layout (~51.3 MB): h | agg | g
  float* h   = (float*)d_ws;
  float* agg = h + (size_t)NN * DH;
  float* g   = agg + (size_t)NN * DH;

  pre_mp_kernel<<<NN / 4, 256, 0, stream>>>(x, Wpre, bpre, h);

  for (int l = 0; l < 3; ++l) {
    zero_f4<<<(NN * DH / 4 + 255) / 256, 256, 0, stream>>>((float4*)agg, NN * DH / 4);
    scatter_add_kernel<<<(NE * 16) / 256, 256, 0, stream>>>(h, ei, agg);
    gin_mlp_kernel<<<NN / 16, 128, 0, stream>>>(h, agg,
        Wc1 + (size_t)l * DH * DH, bc1 + (size_t)l * DH,
        Wc2 + (size_t)l * DH * DH, bc2 + (size_t)l * DH);
  }

  zero_f4<<<(NG * DH / 4 + 255) / 256, 256, 0, stream>>>((float4*)g, NG * DH / 4);
  pool_kernel<<<(int)(((long long)NN * DH + 255) / 256), 256, 0, stream>>>(h, batch, g);
  post_mp_kernel<<<NG, DH, 0, stream>>>(g, Wp1, bp1, Wp2, bp2, out);
}